// MeanPooling_15994458210503
// MI455X (gfx1250) — compile-verified
//
#include <hip/hip_runtime.h>
#include <hip/hip_bf16.h>

#define NUM_GRAPHS 8192
#define DIM 128

typedef __attribute__((ext_vector_type(2))) float     v2f;
typedef __attribute__((ext_vector_type(8))) float     v8f;
typedef __attribute__((ext_vector_type(2))) long long v2i64;

// One workgroup (256 threads = 8 wave32) per segment (batch is sorted, so a
// segment is a contiguous row range found by binary search -> no atomics).
// Wave w owns columns [16w, 16w+16). Rows are summed 4-at-a-time with
// V_WMMA_F32_16X16X4_F32 using an all-ones A matrix:
//   D[m,n] = sum_k B[k,n] + C[m,n]   (exact fp32, RNE)
// so the per-lane K placement in B is irrelevant; only N-striping matters
// (N = lane & 15 per the 32-bit C/D layout). Two accumulators break the
// D->C dependency chain between consecutive WMMAs, letting the XDL pipe
// co-execute with the load stream.
//
// Inner loop uses a single running per-lane pointer (advances 8 rows = 4096 B
// per iteration); the four row loads fold into immediate IOFFSETs so the loop
// body is ~1 VALU add + 4 NT loads (one clause) + 2 WMMAs.
__global__ __launch_bounds__(256) void seg_mean_wmma(
    const float* __restrict__ x, const long long* __restrict__ batch,
    float* __restrict__ emb, float* __restrict__ inv_cnt, int N)
{
    const int g = blockIdx.x;

    __shared__ int s_lo, s_hi;
    if (threadIdx.x == 0) {
        // lower_bound(batch, g) and upper_bound(batch, g) on sorted int64 batch
        int lo = 0, hi = N;
        while (lo < hi) { int m = (lo + hi) >> 1; if (batch[m] < (long long)g) lo = m + 1; else hi = m; }
        s_lo = lo;
        hi = N;
        while (lo < hi) { int m = (lo + hi) >> 1; if (batch[m] <= (long long)g) lo = m + 1; else hi = m; }
        s_hi = lo;
    }
    __syncthreads();
    const int start = s_lo, end = s_hi;
    const int cnt = end - start;

    const int lane = threadIdx.x & 31;
    const int wave = threadIdx.x >> 5;
    const int col  = (wave << 4) + (lane & 15);  // this lane's column (0..127)
    const int rsel = lane >> 4;                  // lane-half -> row slot in B

    const v2f onesA = {1.0f, 1.0f};
    v8f acc0 = {};
    v8f acc1 = {};

    // running per-lane pointer at row (start + rsel), column `col`
    const float* p = x + (size_t)(start + rsel) * DIM + col;

    int r = start;
    // main loop: 8 rows per iteration, 2 independent WMMA chains
    for (; r + 8 <= end; r += 8, p += 8 * DIM) {
        v2f b0, b1;
        b0.x = __builtin_nontemporal_load(p);            // rows r+0 / r+1
        b0.y = __builtin_nontemporal_load(p + 2 * DIM);  // rows r+2 / r+3
        b1.x = __builtin_nontemporal_load(p + 4 * DIM);  // rows r+4 / r+5
        b1.y = __builtin_nontemporal_load(p + 6 * DIM);  // rows r+6 / r+7
        // speculative prefetch of the next 8-row tile (safe past end: dropped)
        __builtin_prefetch(p + 8 * DIM, 0, 0);
        acc0 = __builtin_amdgcn_wmma_f32_16x16x4_f32(
                   false, onesA, false, b0, (short)0, acc0, false, false);
        acc1 = __builtin_amdgcn_wmma_f32_16x16x4_f32(
                   false, onesA, false, b1, (short)0, acc1, false, false);
    }
    // tail: up to 7 rows, masked 4-row chunks (missing rows contribute 0)
    for (; r < end; r += 4, p += 4 * DIM) {
        v2f b;
        b.x = (r + rsel     < end) ? __builtin_nontemporal_load(p)           : 0.0f;
        b.y = (r + 2 + rsel < end) ? __builtin_nontemporal_load(p + 2 * DIM) : 0.0f;
        acc0 = __builtin_amdgcn_wmma_f32_16x16x4_f32(
                   false, onesA, false, b, (short)0, acc0, false, false);
    }

    const float inv = (cnt > 0) ? (1.0f / (float)cnt) : 1.0f;
    // every lane's acc[0] holds the column sum for `col`; lanes 16-31 are
    // duplicates of lanes 0-15 -> only the low half stores.
    if (lane < 16)
        emb[(size_t)g * DIM + col] = (acc0[0] + acc1[0]) * inv;
    if (threadIdx.x == 0)
        inv_cnt[g] = inv;
}

// attention_scores[n] = 1 / max(count[batch[n]], 1): gather from the 32KB
// inv_cnt table (L2-resident; batch sorted -> consecutive threads hit the
// same line). 2 nodes per thread: one b128 index load + one NT b64 store.
__global__ __launch_bounds__(256) void attn_gather(
    const long long* __restrict__ batch, const float* __restrict__ inv_cnt,
    float* __restrict__ attn, int N)
{
    const int i = (blockIdx.x * 256 + threadIdx.x) * 2;
    if (i + 1 < N) {
        const v2i64 b = *reinterpret_cast<const v2i64*>(batch + i);
        v2f o;
        o.x = inv_cnt[(int)b.x];
        o.y = inv_cnt[(int)b.y];
        __builtin_nontemporal_store(o, reinterpret_cast<v2f*>(attn + i));
    } else if (i < N) {
        attn[i] = inv_cnt[(int)batch[i]];
    }
}

extern "C" void kernel_launch(void* const* d_in, const int* in_sizes, int n_in,
                              void* d_out, int out_size, void* d_ws, size_t ws_size,
                              hipStream_t stream)
{
    const float*     x     = (const float*)d_in[0];       // [N, 128] fp32
    const long long* batch = (const long long*)d_in[1];   // [N] int64, sorted
    const int N = in_sizes[1];

    float* emb     = (float*)d_out;                       // [8192, 128]
    float* attn    = emb + (size_t)NUM_GRAPHS * DIM;      // [N]
    float* inv_cnt = (float*)d_ws;                        // [8192] scratch

    seg_mean_wmma<<<NUM_GRAPHS, 256, 0, stream>>>(x, batch, emb, inv_cnt, N);

    const int nb = (N / 2 + 255) / 256;
    attn_gather<<<nb, 256, 0, stream>>>(batch, inv_cnt, attn, N);
}